// PINN_g_rr_32719060861325
// MI455X (gfx1250) — compile-verified
//
#include <hip/hip_runtime.h>
#include <hip/hip_bf16.h>

typedef __attribute__((ext_vector_type(2))) float v2f;
typedef __attribute__((ext_vector_type(8))) float v8f;

#define HID 64
#define RS 66                 // staging row stride (floats), even -> 8B-aligned float2
#define WAVES_PER_BLOCK 4
#define THREADS (WAVES_PER_BLOCK * 32)
#define EBATCH 16             // samples per wave between Einstein passes
#define EST 20                // per-sample floats in epilogue LDS buffer (16B-aligned stride)

// branch-free tanh: v_mul + v_exp_f32 + v_add + v_rcp_f32 + v_fma  (~2-3 ulp)
// saturates correctly: x>>0 -> exp=inf -> rcp=0 -> 1 ; x<<0 -> exp=0 -> 1-2 = -1
__device__ __forceinline__ float fast_tanh(float x) {
    float e = __expf(2.f * x);
    return 1.f - 2.f * __builtin_amdgcn_rcpf(e + 1.f);
}

__global__ __launch_bounds__(THREADS) void pinn_einstein_kernel(
    const float* __restrict__ coords,
    const float* __restrict__ gW1, const float* __restrict__ gB1,
    const float* __restrict__ gW2, const float* __restrict__ gB2,
    const float* __restrict__ gW3, const float* __restrict__ gB3,
    const float* __restrict__ gW4, const float* __restrict__ gB4,
    float* __restrict__ out, int nB)
{
    __shared__ float s_w1[4 * HID];
    __shared__ float s_b1[HID], s_b2[HID], s_b3[HID];
    __shared__ float s_w2t[HID * HID];   // transposed: WT[h][k] = W[k][h]
    __shared__ float s_w3t[HID * HID];
    __shared__ float s_w4[HID];
    __shared__ float s_b4;
    __shared__ float s_stage[WAVES_PER_BLOCK][16 * RS];
    __shared__ float s_est[WAVES_PER_BLOCK][EBATCH * EST];

    const int tid = threadIdx.x;
    for (int i = tid; i < HID * HID; i += THREADS) {
        int h = i >> 6, k = i & 63;
        s_w2t[i] = gW2[k * HID + h];
        s_w3t[i] = gW3[k * HID + h];
    }
    for (int i = tid; i < 4 * HID; i += THREADS) s_w1[i] = gW1[i];
    for (int i = tid; i < HID; i += THREADS) {
        s_b1[i] = gB1[i]; s_b2[i] = gB2[i]; s_b3[i] = gB3[i]; s_w4[i] = gW4[i];
    }
    if (tid == 0) s_b4 = gB4[0];
    __syncthreads();

    const int lane  = tid & 31;
    const int wid   = tid >> 5;
    const bool low  = lane < 16;
    const int col16 = lane & 15;
    const int koff  = low ? 0 : 2;   // K offset of this half-wave inside a 16x16x4 fragment
    float* stage  = &s_stage[wid][0];
    float* estage = &s_est[wid][0];

    const int waveGlobal = blockIdx.x * WAVES_PER_BLOCK + wid;
    const int waveStride = gridDim.x * WAVES_PER_BLOCK;

    for (int s0 = waveGlobal * EBATCH; s0 < nB; s0 += waveStride * EBATCH) {

        // ================== MLP + forward-mode AD for EBATCH samples ==================
        for (int sub = 0; sub < EBATCH; ++sub) {
            int s = s0 + sub;
            if (s >= nB) break;                       // wave-uniform
            const float4 c = reinterpret_cast<const float4*>(coords)[s];

            // ---- Layer 1 (4 -> 64), analytic fwd-mode seed ----
            {
                int h0 = lane * 2;
                float rv[16][2];
                #pragma unroll
                for (int u = 0; u < 2; ++u) {
                    int h = h0 + u;
                    float w0 = s_w1[0 * HID + h], w1 = s_w1[1 * HID + h];
                    float w2 = s_w1[2 * HID + h], w3 = s_w1[3 * HID + h];
                    float z = c.x * w0 + c.y * w1 + c.z * w2 + c.w * w3 + s_b1[h];
                    float v = fast_tanh(z);
                    float w = 1.f - v * v;
                    float m = -2.f * v * w;
                    rv[0][u] = v;
                    rv[1][u] = w * w0; rv[2][u] = w * w1; rv[3][u] = w * w2; rv[4][u] = w * w3;
                    rv[5][u]  = m * w0 * w0; rv[6][u]  = m * w0 * w1; rv[7][u]  = m * w0 * w2;
                    rv[8][u]  = m * w0 * w3; rv[9][u]  = m * w1 * w1; rv[10][u] = m * w1 * w2;
                    rv[11][u] = m * w1 * w3; rv[12][u] = m * w2 * w2; rv[13][u] = m * w2 * w3;
                    rv[14][u] = m * w3 * w3; rv[15][u] = 0.f;
                }
                #pragma unroll
                for (int r = 0; r < 16; ++r) {
                    v2f pv; pv.x = rv[r][0]; pv.y = rv[r][1];
                    *(v2f*)(stage + r * RS + h0) = pv;
                }
            }

            // ---- Layers 2 & 3 (64 -> 64) via f32 WMMA, 4 independent chains ----
            float n3[4][8];   // layer-3 outputs kept in D-layout registers
            #pragma unroll
            for (int layer = 0; layer < 2; ++layer) {
                const float* wt = (layer == 0) ? s_w2t : s_w3t;
                const float* bv = (layer == 0) ? s_b2  : s_b3;

                v2f afrag[16];
                const float* arow = stage + col16 * RS + koff;
                #pragma unroll
                for (int kk = 0; kk < 16; ++kk) afrag[kk] = *(const v2f*)(arow + 4 * kk);

                const float* brow = wt + col16 * HID + koff;
                v8f acc[4];
                #pragma unroll
                for (int t = 0; t < 4; ++t) acc[t] = (v8f){};

                // k-outer / tile-inner: consecutive WMMAs hit different accumulators
                #pragma unroll
                for (int kk = 0; kk < 16; ++kk) {
                    v2f bf[4];
                    #pragma unroll
                    for (int t = 0; t < 4; ++t)
                        bf[t] = *(const v2f*)(brow + t * 16 * HID + 4 * kk);
                    #pragma unroll
                    for (int t = 0; t < 4; ++t)
                        acc[t] = __builtin_amdgcn_wmma_f32_16x16x4_f32(
                            false, afrag[kk], false, bf[t], (short)0, acc[t], false, false);
                }

                #pragma unroll
                for (int t = 0; t < 4; ++t) {
                    int h = t * 16 + col16;
                    float cv[8];
                    #pragma unroll
                    for (int r = 0; r < 8; ++r) cv[r] = acc[t][r];

                    float v0 = fast_tanh(cv[0] + bv[h]);       // valid in low half-lane
                    float d0l = cv[1], d1l = cv[2], d2l = cv[3], d3l = cv[4];
                    float vX  = __shfl_xor(v0, 16, 32);
                    float d0X = __shfl_xor(d0l, 16, 32);
                    float d1X = __shfl_xor(d1l, 16, 32);
                    float d2X = __shfl_xor(d2l, 16, 32);
                    float d3X = __shfl_xor(d3l, 16, 32);
                    float v  = low ? v0  : vX;
                    float d0 = low ? d0l : d0X;
                    float d1 = low ? d1l : d1X;
                    float d2 = low ? d2l : d2X;
                    float d3 = low ? d3l : d3X;
                    float w  = 1.f - v * v;
                    float tv = 2.f * v;

                    float nn[8];
                    if (low) {  // rows 0..7: v, J0..J3, H(0,0) H(0,1) H(0,2)
                        nn[0] = v;
                        nn[1] = w * d0; nn[2] = w * d1; nn[3] = w * d2; nn[4] = w * d3;
                        nn[5] = w * (cv[5] - tv * d0 * d0);
                        nn[6] = w * (cv[6] - tv * d0 * d1);
                        nn[7] = w * (cv[7] - tv * d0 * d2);
                    } else {    // rows 8..15: H(0,3)(1,1)(1,2)(1,3)(2,2)(2,3)(3,3), pad
                        nn[0] = w * (cv[0] - tv * d0 * d3);
                        nn[1] = w * (cv[1] - tv * d1 * d1);
                        nn[2] = w * (cv[2] - tv * d1 * d2);
                        nn[3] = w * (cv[3] - tv * d1 * d3);
                        nn[4] = w * (cv[4] - tv * d2 * d2);
                        nn[5] = w * (cv[5] - tv * d2 * d3);
                        nn[6] = w * (cv[6] - tv * d3 * d3);
                        nn[7] = 0.f;
                    }

                    if (layer == 0) {
                        #pragma unroll
                        for (int r = 0; r < 8; ++r)
                            stage[(low ? r : r + 8) * RS + h] = nn[r];
                    } else {
                        #pragma unroll
                        for (int r = 0; r < 8; ++r) n3[t][r] = nn[r];
                    }
                }
            }

            // ---- Layer 4 (64 -> 1): f, grad(4), hess(10) via butterfly reduce ----
            float a15[15];
            #pragma unroll
            for (int r = 0; r < 15; ++r) a15[r] = 0.f;
            #pragma unroll
            for (int t = 0; t < 4; ++t) {
                float wv = s_w4[t * 16 + col16];
                if (low) {
                    #pragma unroll
                    for (int r = 0; r < 8; ++r) a15[r] += n3[t][r] * wv;
                } else {
                    #pragma unroll
                    for (int r = 0; r < 7; ++r) a15[8 + r] += n3[t][r] * wv;
                }
            }
            #pragma unroll
            for (int m = 1; m < 32; m <<= 1) {
                #pragma unroll
                for (int r = 0; r < 15; ++r) a15[r] += __shfl_xor(a15[r], m, 32);
            }

            // park this sample's (f, grad, hess) in LDS for the batched epilogue
            if (lane == 0) {
                float* ep = estage + sub * EST;
                float4 p0; p0.x = a15[0];  p0.y = a15[1];  p0.z = a15[2];  p0.w = a15[3];
                float4 p1; p1.x = a15[4];  p1.y = a15[5];  p1.z = a15[6];  p1.w = a15[7];
                float4 p2; p2.x = a15[8];  p2.y = a15[9];  p2.z = a15[10]; p2.w = a15[11];
                *(float4*)(ep + 0) = p0;
                *(float4*)(ep + 4) = p1;
                *(float4*)(ep + 8) = p2;
                v2f p3; p3.x = a15[12]; p3.y = a15[13];
                *(v2f*)(ep + 12) = p3;
                ep[14] = a15[14];
            }
        }

        // ================== Einstein tensor: lane-parallel, lane = sample ==================
        {
            int sE = s0 + lane;
            if (low && sE < nB) {
                const float* ep = estage + lane * EST;
                float4 p0 = *(const float4*)(ep + 0);
                float4 p1 = *(const float4*)(ep + 4);
                float4 p2 = *(const float4*)(ep + 8);
                v2f    p3 = *(const v2f*)(ep + 12);
                float  p4 = ep[14];

                float fval = p0.x + s_b4;
                float fd[4] = { p0.y, p0.z, p0.w, p1.x };
                float fdd[4][4];
                fdd[0][0] = p1.y;
                fdd[0][1] = fdd[1][0] = p1.z;
                fdd[0][2] = fdd[2][0] = p1.w;
                fdd[0][3] = fdd[3][0] = p2.x;
                fdd[1][1] = p2.y;
                fdd[1][2] = fdd[2][1] = p2.z;
                fdd[1][3] = fdd[3][1] = p2.w;
                fdd[2][2] = p3.x;
                fdd[2][3] = fdd[3][2] = p3.y;
                fdd[3][3] = p4;

                const float4 c = reinterpret_cast<const float4*>(coords)[sE];
                float Ae = expf(fval);
                float r_ = c.y, th = c.z;
                float sn = sinf(th), cs = cosf(th);
                float r2 = r_ * r_, s2 = sn * sn;
                float gi[4] = { -1.f, 1.f / Ae, 1.f / r2, 1.f / (r2 * s2) };

                float Dm[4][4];                       // Dm[i][j] = d_j g_ii
                #pragma unroll
                for (int i = 0; i < 4; ++i)
                    #pragma unroll
                    for (int j = 0; j < 4; ++j) Dm[i][j] = 0.f;
                #pragma unroll
                for (int j = 0; j < 4; ++j) Dm[1][j] = Ae * fd[j];
                Dm[2][1] = 2.f * r_;
                Dm[3][1] = 2.f * r_ * s2;
                Dm[3][2] = 2.f * r2 * sn * cs;

                float DDm[4][4][4];                   // DDm[i][j][l] = d_l d_j g_ii
                #pragma unroll
                for (int i = 0; i < 4; ++i)
                    #pragma unroll
                    for (int j = 0; j < 4; ++j)
                        #pragma unroll
                        for (int l = 0; l < 4; ++l) DDm[i][j][l] = 0.f;
                #pragma unroll
                for (int j = 0; j < 4; ++j)
                    #pragma unroll
                    for (int l = 0; l < 4; ++l)
                        DDm[1][j][l] = Ae * (fdd[j][l] + fd[j] * fd[l]);
                DDm[2][1][1] = 2.f;
                DDm[3][1][1] = 2.f * s2;
                DDm[3][1][2] = DDm[3][2][1] = 4.f * r_ * sn * cs;
                DDm[3][2][2] = 2.f * r2 * (cs * cs - s2);

                float Gm[4][4][4];                    // Gamma^i_{kj}
                #pragma unroll
                for (int i = 0; i < 4; ++i)
                    #pragma unroll
                    for (int k = 0; k < 4; ++k)
                        #pragma unroll
                        for (int j = 0; j < 4; ++j) {
                            float T = 0.f;
                            if (i == k) T += Dm[i][j];
                            if (i == j) T += Dm[i][k];
                            if (k == j) T -= Dm[k][i];
                            Gm[i][k][j] = 0.5f * gi[i] * T;
                        }

                float trG[4];
                #pragma unroll
                for (int m = 0; m < 4; ++m)
                    trG[m] = Gm[0][m][0] + Gm[1][m][1] + Gm[2][m][2] + Gm[3][m][3];

                auto dGam = [&](int i, int k, int j, int l) -> float { // d_l Gamma^i_{kj}
                    float T = 0.f, dT = 0.f;
                    if (i == k) { T += Dm[i][j]; dT += DDm[i][j][l]; }
                    if (i == j) { T += Dm[i][k]; dT += DDm[i][k][l]; }
                    if (k == j) { T -= Dm[k][i]; dT -= DDm[k][i][l]; }
                    return -0.5f * gi[i] * gi[i] * Dm[i][l] * T + 0.5f * gi[i] * dT;
                };

                float ric[4][4];
                #pragma unroll
                for (int p = 0; p < 4; ++p)
                    #pragma unroll
                    for (int q = 0; q < 4; ++q) {
                        float a = 0.f;
                        #pragma unroll
                        for (int m = 0; m < 4; ++m) a += Gm[m][p][q] * trG[m];
                        #pragma unroll
                        for (int i = 0; i < 4; ++i)
                            #pragma unroll
                            for (int m = 0; m < 4; ++m) a -= Gm[m][p][i] * Gm[i][m][q];
                        #pragma unroll
                        for (int i = 0; i < 4; ++i) a += dGam(i, p, q, i) - dGam(i, p, i, q);
                        ric[p][q] = a;
                    }

                float Rs = gi[0] * ric[0][0] + gi[1] * ric[1][1]
                         + gi[2] * ric[2][2] + gi[3] * ric[3][3];

                float* op = out + (size_t)sE * 16;
                #pragma unroll
                for (int p = 0; p < 4; ++p) {
                    float4 ov;
                    ov.x = gi[p] * gi[0] * ric[p][0] - ((p == 0) ? 0.5f * Rs * gi[p] : 0.f);
                    ov.y = gi[p] * gi[1] * ric[p][1] - ((p == 1) ? 0.5f * Rs * gi[p] : 0.f);
                    ov.z = gi[p] * gi[2] * ric[p][2] - ((p == 2) ? 0.5f * Rs * gi[p] : 0.f);
                    ov.w = gi[p] * gi[3] * ric[p][3] - ((p == 3) ? 0.5f * Rs * gi[p] : 0.f);
                    *(float4*)(op + p * 4) = ov;
                }
            }
        }
    }
}

extern "C" void kernel_launch(void* const* d_in, const int* in_sizes, int n_in,
                              void* d_out, int out_size, void* d_ws, size_t ws_size,
                              hipStream_t stream) {
    (void)n_in; (void)out_size; (void)d_ws; (void)ws_size;
    const float* coords = (const float*)d_in[0];
    const float* W1 = (const float*)d_in[1];
    const float* b1 = (const float*)d_in[2];
    const float* W2 = (const float*)d_in[3];
    const float* b2 = (const float*)d_in[4];
    const float* W3 = (const float*)d_in[5];
    const float* b3 = (const float*)d_in[6];
    const float* W4 = (const float*)d_in[7];
    const float* b4 = (const float*)d_in[8];
    float* out = (float*)d_out;

    int nB = in_sizes[0] / 4;
    int perBlock = WAVES_PER_BLOCK * EBATCH;          // 64 samples per block
    int blocks = (nB + perBlock - 1) / perBlock;
    if (blocks < 1) blocks = 1;

    pinn_einstein_kernel<<<blocks, THREADS, 0, stream>>>(
        coords, W1, b1, W2, b2, W3, b3, W4, b4, out, nB);
}